// Net_16793322127646
// MI455X (gfx1250) — compile-verified
//
#include <hip/hip_runtime.h>
#include <hip/hip_bf16.h>

// ---------------------------------------------------------------------------
// Problem constants (from the reference)
// ---------------------------------------------------------------------------
#define BB    16          // batch
#define TT    2048        // timesteps
#define CIN   17          // conv in channels
#define HW    8           // input H=W
#define NCONV 4           // conv out channels
#define INSZ  144         // 4*6*6 features after conv
#define HID   150         // LSTM hidden
#define NG    600         // 4*HID gates
#define NGP   608         // gates padded to 16
#define NTILE 38          // 608/16 N-tiles
#define KI    36          // 144/4 k-steps (input GEMM, exact)
#define KQ    38          // ceil(150/4) k-steps (recurrent GEMM, padded to 152)
#define HIDP  152         // hidden padded to multiple of 4

typedef float v8f __attribute__((ext_vector_type(8)));
typedef float v2f __attribute__((ext_vector_type(2)));

// D = A(16x4 f32) * B(4x16 f32) + C(16x16 f32), full-precision WMMA.
__device__ __forceinline__ v8f wmma_f32_k4(v2f a, v2f b, v8f c) {
    // 8-arg form: (neg_a, A, neg_b, B, c_mod, C, reuse_a, reuse_b)
    return __builtin_amdgcn_wmma_f32_16x16x4_f32(false, a, false, b,
                                                 (short)0, c, false, false);
}

__device__ __forceinline__ float sigmoidf_(float x) {
    return 1.0f / (1.0f + __expf(-x));
}

// ---------------------------------------------------------------------------
// Kernel 1: 3x3 VALID conv + bias + ReLU  -> y[s][144], s = b*T + t
// Memory-bound phase: 143 MB input @ 23.3 TB/s is the floor; LDS-stage each
// sample once (1088 floats) so every input byte is read from HBM exactly once.
// ---------------------------------------------------------------------------
__global__ __launch_bounds__(256) void conv_relu_kernel(
    const float* __restrict__ in, const float* __restrict__ cw,
    const float* __restrict__ cb, float* __restrict__ y) {
    __shared__ float s_in[4][CIN * HW * HW];   // 4 samples * 1088 floats
    __shared__ float s_w[NCONV * CIN * 9];
    __shared__ float s_b[NCONV];
    const int tid = threadIdx.x;
    const long long s0 = (long long)blockIdx.x * 4;

    for (int i = tid; i < NCONV * CIN * 9; i += 256) s_w[i] = cw[i];
    if (tid < NCONV) s_b[tid] = cb[tid];
    for (int i = tid; i < 4 * CIN * HW * HW; i += 256) {
        int si = i / (CIN * HW * HW);
        int off = i % (CIN * HW * HW);
        s_in[si][off] = in[(s0 + si) * (CIN * HW * HW) + off];
    }
    __syncthreads();

    for (int i = tid; i < 4 * INSZ; i += 256) {
        int si = i / INSZ;
        int f  = i % INSZ;                 // f = oc*36 + oy*6 + ox (matches reshape)
        int oc = f / 36;
        int oy = (f % 36) / 6;
        int ox = f % 6;
        float acc = s_b[oc];
        const float* w  = &s_w[oc * CIN * 9];
        const float* sp = &s_in[si][0];
        #pragma unroll
        for (int c = 0; c < CIN; ++c)
            #pragma unroll
            for (int ky = 0; ky < 3; ++ky)
                #pragma unroll
                for (int kx = 0; kx < 3; ++kx)
                    acc = fmaf(sp[c * 64 + (oy + ky) * 8 + (ox + kx)],
                               w[c * 9 + ky * 3 + kx], acc);
        y[(s0 + si) * INSZ + f] = fmaxf(acc, 0.0f);
    }
}

// ---------------------------------------------------------------------------
// Kernel 2: gates_x = y[32768x144] @ w_ih.T[144x600] + (b_ih + b_hh)
// f32 WMMA 16x16x4. Each wave owns one 16-wide N-tile; B (w_ih tile) is held
// in 72 VGPRs and reused across 32 M-tiles; the A-tile is staged transposed
// in LDS. Accumulation split into two independent WMMA chains (even/odd k)
// so the loop runs at issue rate, not at WMMA latency.
// Output written time-major: gx[t][b][gate].
// ---------------------------------------------------------------------------
__global__ __launch_bounds__(256) void gemm_input_kernel(
    const float* __restrict__ y, const float* __restrict__ w_ih,
    const float* __restrict__ b_ih, const float* __restrict__ b_hh,
    float* __restrict__ gx) {
    __shared__ float s_yT[INSZ * 16];      // [f][row]
    const int tid  = threadIdx.x;
    const int lane = tid & 31;
    const int wid  = tid >> 5;                       // 0..7
    const int tile = blockIdx.y * 8 + wid;           // N-tile id
    const int c0   = (lane < 16) ? 0 : 2;            // K sub-offset per lane half
    const int bl   = lane & 15;
    const int gate = tile * 16 + bl;
    const bool act = (tile < NTILE) && (gate < NG);

    v2f Breg[KI];
    #pragma unroll
    for (int ks = 0; ks < KI; ++ks) {
        const int k0 = 4 * ks + c0;
        float bx = 0.0f, by = 0.0f;
        if (act) { bx = w_ih[gate * INSZ + k0]; by = w_ih[gate * INSZ + k0 + 1]; }
        Breg[ks].x = bx; Breg[ks].y = by;
    }
    const float bias = act ? (b_ih[gate] + b_hh[gate]) : 0.0f;

    const int M_PER_BLOCK = (BB * TT / 16) / 64;     // 32 M-tiles per block
    for (int mi = 0; mi < M_PER_BLOCK; ++mi) {
        const long long m0 = ((long long)blockIdx.x * M_PER_BLOCK + mi) * 16;
        __syncthreads();
        for (int i = tid; i < 16 * INSZ; i += 256) {
            int r = i / INSZ, f = i % INSZ;
            s_yT[f * 16 + r] = y[(m0 + r) * INSZ + f];
        }
        __syncthreads();
        if (tile < NTILE) {
            v8f acc0 = {}, acc1 = {};                // two independent chains
            #pragma unroll
            for (int ks = 0; ks < KI; ks += 2) {
                const int k0 = 4 * ks + c0;
                v2f a0, a1;
                a0.x = s_yT[k0 * 16 + bl];
                a0.y = s_yT[(k0 + 1) * 16 + bl];
                a1.x = s_yT[(k0 + 4) * 16 + bl];
                a1.y = s_yT[(k0 + 5) * 16 + bl];
                acc0 = wmma_f32_k4(a0, Breg[ks],     acc0);
                acc1 = wmma_f32_k4(a1, Breg[ks + 1], acc1);
            }
            if (gate < NG) {
                #pragma unroll
                for (int r = 0; r < 8; ++r) {
                    const int  M = r + ((lane < 16) ? 0 : 8);
                    const long long s = m0 + M;
                    const long long b = s >> 11;     // s / TT
                    const long long t = s & (TT - 1);
                    gx[(t * BB + b) * NG + gate] = acc0[r] + acc1[r] + bias;
                }
            }
        }
    }
}

// ---------------------------------------------------------------------------
// Kernel 3: the sequential LSTM. One persistent workgroup (19 waves / 608
// threads) on one WGP; h, c, and the gate buffer live in LDS; w_hh (352 KB,
// too big for 320 KB LDS) is register-resident: 19 waves x 2 tiles x 38
// k-slices x 2 VGPRs == padded 608x152 matrix. Per step: 4 independent WMMA
// chains (2 tiles x even/odd k), gx preloaded into registers so its L2
// latency hides behind the WMMAs, barrier, fused gate math, barrier.
// ---------------------------------------------------------------------------
__global__ __launch_bounds__(608, 1) void lstm_kernel(
    const float* __restrict__ gx, const float* __restrict__ w_hh,
    const float* __restrict__ last_w, const float* __restrict__ last_b,
    float* __restrict__ out) {
    __shared__ float s_h[HIDP * 16];   // h transposed: [hid][b], rows 150/151 stay 0
    __shared__ float s_c[HID * 16];    // c transposed: [hid][b]
    __shared__ float s_g[16 * NGP];    // raw recurrent GEMM result: [b][gate]

    const int tid  = threadIdx.x;
    const int lane = tid & 31;
    const int wid  = tid >> 5;                  // 0..18
    const int c0   = (lane < 16) ? 0 : 2;
    const int bl   = lane & 15;
    const int tA = wid * 2, tB = wid * 2 + 1;   // this wave's two N-tiles
    const int gA = tA * 16 + bl, gB = tB * 16 + bl;

    // Register-resident B operands (w_hh.T tiles), zero-padded at the edges.
    v2f BA[KQ], BBv[KQ];
    #pragma unroll
    for (int ks = 0; ks < KQ; ++ks) {
        const int k0 = 4 * ks + c0;
        float ax = 0.0f, ay = 0.0f, bx = 0.0f, by = 0.0f;
        if (gA < NG) {
            if (k0     < HID) ax = w_hh[gA * HID + k0];
            if (k0 + 1 < HID) ay = w_hh[gA * HID + k0 + 1];
        }
        if (gB < NG) {
            if (k0     < HID) bx = w_hh[gB * HID + k0];
            if (k0 + 1 < HID) by = w_hh[gB * HID + k0 + 1];
        }
        BA[ks].x = ax;  BA[ks].y = ay;
        BBv[ks].x = bx; BBv[ks].y = by;
    }

    // Hoisted index maps for the elementwise phase (loop-invariant over t).
    int  off_g[4], off_c[4], off_x[4];
    bool val4[4];
    #pragma unroll
    for (int j = 0; j < 4; ++j) {
        const int i = tid + j * 608;
        const bool v = (i < BB * HID);
        const int b   = v ? (i / HID) : 0;
        const int hid = v ? (i % HID) : 0;
        val4[j]  = v;
        off_g[j] = b * NGP + hid;
        off_c[j] = hid * 16 + b;
        off_x[j] = b * NG + hid;
    }

    for (int i = tid; i < HIDP * 16; i += 608) s_h[i] = 0.0f;
    for (int i = tid; i < HID * 16;  i += 608) s_c[i] = 0.0f;
    __syncthreads();

    const float* gxt = gx;
    for (int t = 0; t < TT; ++t) {
        // Prefetch next step's gates_x slab (608 lanes x 64B covers 38.4 KB).
        if (t + 1 < TT) {
            const char* nxt = (const char*)(gxt + BB * NG);
            __builtin_prefetch(nxt + (size_t)tid * 64, 0, 1);
        }
        // Preload this step's gx operands; loads stay in flight during WMMAs.
        float pgi[4], pgf[4], pgg[4], pgo[4];
        #pragma unroll
        for (int j = 0; j < 4; ++j) {
            if (val4[j]) {
                const float* p = gxt + off_x[j];
                pgi[j] = p[0];
                pgf[j] = p[150];
                pgg[j] = p[300];
                pgo[j] = p[450];
            }
        }

        // ---- recurrent GEMM: s_g = h[16x150] @ w_hh.T[150x600] ----
        v8f aA0 = {}, aA1 = {}, aB0 = {}, aB1 = {};   // 4 independent chains
        #pragma unroll
        for (int ks = 0; ks < KQ; ks += 2) {
            const int k0 = 4 * ks + c0;
            v2f a0, a1;
            a0.x = s_h[k0 * 16 + bl];
            a0.y = s_h[(k0 + 1) * 16 + bl];
            a1.x = s_h[(k0 + 4) * 16 + bl];
            a1.y = s_h[(k0 + 5) * 16 + bl];
            aA0 = wmma_f32_k4(a0, BA[ks],      aA0);
            aB0 = wmma_f32_k4(a0, BBv[ks],     aB0);
            aA1 = wmma_f32_k4(a1, BA[ks + 1],  aA1);
            aB1 = wmma_f32_k4(a1, BBv[ks + 1], aB1);
        }
        #pragma unroll
        for (int r = 0; r < 8; ++r) {
            const int M = r + ((lane < 16) ? 0 : 8);
            s_g[M * NGP + tA * 16 + bl] = aA0[r] + aA1[r];
            s_g[M * NGP + tB * 16 + bl] = aB0[r] + aB1[r];
        }
        __syncthreads();

        // ---- fused gate nonlinearities + state update ----
        #pragma unroll
        for (int j = 0; j < 4; ++j) {
            if (val4[j]) {
                const float gi = s_g[off_g[j]]       + pgi[j];
                const float gf = s_g[off_g[j] + 150] + pgf[j];
                const float gg = s_g[off_g[j] + 300] + pgg[j];
                const float go = s_g[off_g[j] + 450] + pgo[j];
                const float cc = sigmoidf_(gf) * s_c[off_c[j]]
                               + sigmoidf_(gi) * tanhf(gg);
                s_c[off_c[j]] = cc;
                s_h[off_c[j]] = sigmoidf_(go) * tanhf(cc);
            }
        }
        __syncthreads();
        gxt += BB * NG;
    }

    // ---- final projection: out[b][o] = h_T . last_w[o] + last_b[o] ----
    if (tid < 32) {
        const int b = tid >> 1, o = tid & 1;
        float acc = last_b[o];
        for (int hid = 0; hid < HID; ++hid)
            acc = fmaf(s_h[hid * 16 + b], last_w[o * HID + hid], acc);
        out[b * 2 + o] = acc;
    }
}

// ---------------------------------------------------------------------------
extern "C" void kernel_launch(void* const* d_in, const int* in_sizes, int n_in,
                              void* d_out, int out_size, void* d_ws, size_t ws_size,
                              hipStream_t stream) {
    const float* inputs = (const float*)d_in[0];
    const float* conv_w = (const float*)d_in[1];
    const float* conv_b = (const float*)d_in[2];
    const float* w_ih   = (const float*)d_in[3];
    const float* w_hh   = (const float*)d_in[4];
    const float* b_ih   = (const float*)d_in[5];
    const float* b_hh   = (const float*)d_in[6];
    const float* last_w = (const float*)d_in[7];
    const float* last_b = (const float*)d_in[8];

    float* y  = (float*)d_ws;                                     // 32768*144 f32
    float* gx = (float*)((char*)d_ws +
                         (size_t)BB * TT * INSZ * sizeof(float)); // [T][B][600] f32

    conv_relu_kernel<<<BB * TT / 4, 256, 0, stream>>>(inputs, conv_w, conv_b, y);
    gemm_input_kernel<<<dim3(64, 5), 256, 0, stream>>>(y, w_ih, b_ih, b_hh, gx);
    lstm_kernel<<<1, 608, 0, stream>>>(gx, w_hh, last_w, last_b, (float*)d_out);
}